// ShortConv1d_62242666053983
// MI455X (gfx1250) — compile-verified
//
#include <hip/hip_runtime.h>

// Depthwise causal conv1d (K=4) + bias + SiLU over x:(B,T,C) fp32, C contiguous.
// Pure HBM-streaming kernel: ~268 MB traffic -> ~11.5us floor at 23.3 TB/s.
// Data path: gfx1250 async global->LDS (ASYNCcnt) double-buffered staging,
// register rotation along time for the K-1 halo, NT 128-bit stores.

#define KSZ   4
#define B_    4
#define T_    4096
#define C_    2048
#define TPB   256                 // threads per block (8 wave32)
#define CHB   (TPB * 4)           // 1024 channels per block
#define ROWS  8                   // time rows per LDS tile
#define TILES 16                  // tiles per block
#define TT    (ROWS * TILES)      // 128 time rows per block

static_assert(C_ % CHB == 0, "channel tiling");
static_assert(T_ % TT == 0, "time tiling");

typedef float vf4 __attribute__((ext_vector_type(4)));

// Builtin prototype (from hipcc diagnostic): (int4 AS1*, int4 AS3*, Ii, Ii)
// where int4 is a GCC-style vector_size(16) int.
typedef int v4i __attribute__((vector_size(16)));
typedef __attribute__((address_space(1))) v4i gv4i_t;
typedef __attribute__((address_space(3))) v4i lv4i_t;

#if __has_builtin(__builtin_amdgcn_global_load_async_to_lds_b128)
#define HAVE_ASYNC_B128 1
#endif

#if __has_builtin(__builtin_amdgcn_s_wait_asynccnt)
#define WAIT_ASYNC(n) __builtin_amdgcn_s_wait_asynccnt(n)
#else
#define WAIT_ASYNC(n) asm volatile("s_wait_asynccnt %0" ::"i"(n) : "memory")
#endif

__device__ __forceinline__ void async_cp16(const vf4* g, vf4* l) {
#ifdef HAVE_ASYNC_B128
  __builtin_amdgcn_global_load_async_to_lds_b128(
      (gv4i_t*)g, (lv4i_t*)l, /*offset=*/0, /*cpol=*/0);
#else
  unsigned loff = (unsigned)(unsigned long long)(__attribute__((address_space(3))) vf4*)l;
  asm volatile("global_load_async_to_lds_b128 %0, %1, off"
               ::"v"(loff), "v"(g)
               : "memory");
#endif
}

__device__ __forceinline__ float silu1(float v) {
  // v * sigmoid(v); native exp + rcp, plenty of headroom under the HBM bound
  return v * __builtin_amdgcn_rcpf(1.0f + __expf(-v));
}

__global__ __launch_bounds__(TPB) void ShortConv1d_silu_kernel(
    const float* __restrict__ x, const float* __restrict__ w,
    const float* __restrict__ bias, float* __restrict__ y) {
  __shared__ vf4 smem[2 * ROWS * TPB];  // 64 KB, double-buffered tiles

  const int tid    = threadIdx.x;
  const int cblk   = blockIdx.x;  // 0..C_/CHB-1
  const int tchunk = blockIdx.y;  // 0..T_/TT-1
  const int b      = blockIdx.z;  // 0..B_-1

  const int c0 = cblk * CHB + tid * 4;  // 4 contiguous channels per thread
  const int t0 = tchunk * TT;

  // Per-channel taps: w is (C, K) row-major -> 16 contiguous floats for c0..c0+3
  const vf4 wA = *(const vf4*)(w + (size_t)(c0 + 0) * KSZ);
  const vf4 wB = *(const vf4*)(w + (size_t)(c0 + 1) * KSZ);
  const vf4 wC = *(const vf4*)(w + (size_t)(c0 + 2) * KSZ);
  const vf4 wD = *(const vf4*)(w + (size_t)(c0 + 3) * KSZ);
  const vf4 bs = *(const vf4*)(bias + c0);

  const size_t base = ((size_t)b * T_ + t0) * C_ + c0;
  const vf4* __restrict__ xg = (const vf4*)(x + base);
  vf4* __restrict__ yg       = (vf4*)(y + base);
  const int rs = C_ / 4;  // row stride in vf4 units

  // Halo: x at t0-3, t0-2, t0-1 (zeros for the causal left pad at t0==0)
  vf4 x0, x1, x2;
  if (t0 == 0) {
    x0 = (vf4)0.0f; x1 = (vf4)0.0f; x2 = (vf4)0.0f;
  } else {
    x0 = xg[-3 * rs]; x1 = xg[-2 * rs]; x2 = xg[-1 * rs];
  }

  // Prime the async pipeline: tiles 0 and 1 (8 B128 copies each per thread).
#pragma unroll
  for (int r = 0; r < ROWS; ++r)
    async_cp16(xg + r * rs, &smem[r * TPB + tid]);
#pragma unroll
  for (int r = 0; r < ROWS; ++r)
    async_cp16(xg + (ROWS + r) * rs, &smem[(ROWS + r) * TPB + tid]);

  for (int tile = 0; tile < TILES; ++tile) {
    // Async completion is in-order: <=ROWS pending means this tile landed,
    // while the next tile's ROWS copies remain in flight.
    if (tile + 1 < TILES) { WAIT_ASYNC(ROWS); } else { WAIT_ASYNC(0); }

    const int buf = tile & 1;
    const vf4* sb = &smem[buf * ROWS * TPB];

#pragma unroll
    for (int r = 0; r < ROWS; ++r) {
      const vf4 x3 = sb[r * TPB + tid];  // ds_load_b128, own slot only
      vf4 v;
      v.x = bs.x + wA.x * x0.x + wA.y * x1.x + wA.z * x2.x + wA.w * x3.x;
      v.y = bs.y + wB.x * x0.y + wB.y * x1.y + wB.z * x2.y + wB.w * x3.y;
      v.z = bs.z + wC.x * x0.z + wC.y * x1.z + wC.z * x2.z + wC.w * x3.z;
      v.w = bs.w + wD.x * x0.w + wD.y * x1.w + wD.z * x2.w + wD.w * x3.w;
      v.x = silu1(v.x); v.y = silu1(v.y); v.z = silu1(v.z); v.w = silu1(v.w);
      // Streamed once, never re-read, and x+y (268MB) > L2 (192MB): NT store.
      __builtin_nontemporal_store(v, yg + (tile * ROWS + r) * rs);
      x0 = x1; x1 = x2; x2 = x3;  // rotate causal window
    }

    // Refill the buffer we just drained with tile+2.
    if (tile + 2 < TILES) {
      asm volatile("" ::: "memory");  // keep issue after the ds reads above
      const int rt = (tile + 2) * ROWS;
#pragma unroll
      for (int r = 0; r < ROWS; ++r)
        async_cp16(xg + (rt + r) * rs, &smem[(buf * ROWS + r) * TPB + tid]);
    }
  }
}

extern "C" void kernel_launch(void* const* d_in, const int* in_sizes, int n_in,
                              void* d_out, int out_size, void* d_ws, size_t ws_size,
                              hipStream_t stream) {
  (void)in_sizes; (void)n_in; (void)out_size; (void)d_ws; (void)ws_size;
  const float* x    = (const float*)d_in[0];
  const float* w    = (const float*)d_in[1];
  const float* bias = (const float*)d_in[2];
  float* out        = (float*)d_out;

  dim3 grid(C_ / CHB, T_ / TT, B_);  // (2, 32, 4) = 256 workgroups
  dim3 block(TPB);
  hipLaunchKernelGGL(ShortConv1d_silu_kernel, grid, block, 0, stream, x, w, bias, out);
}